// CGSC_64914135712264
// MI455X (gfx1250) — compile-verified
//
#include <hip/hip_runtime.h>
#include <hip/hip_bf16.h>
#include <math.h>

// ---------------------------------------------------------------------------
// CGSC block forward for MI455X (gfx1250, wave32).
// GEMM parts (1x1 convs) use V_WMMA_F32_16X16X4_F32.
// Dilated-reparam lepe branch is folded into ONE effective 7x7 depthwise conv.
// ---------------------------------------------------------------------------

typedef float v2f __attribute__((ext_vector_type(2)));
typedef float v8f __attribute__((ext_vector_type(8)));

static constexpr int BN   = 4;
static constexpr int CC   = 128;
static constexpr int HH   = 56;
static constexpr int WW   = 56;
static constexpr int NP   = HH * WW;      // 3136
static constexpr int K2   = 49;
static constexpr int RDIM = 8;            // C / 16

// ---------------------------------------------------------------------------
// 1) Build effective 7x7 depthwise kernel from the 5 reparam branches + BNs.
// ---------------------------------------------------------------------------
__global__ void build_eff_kernel(
    const float* __restrict__ lk_w, const float* __restrict__ lk_s, const float* __restrict__ lk_b,
    const float* __restrict__ dw5,  const float* __restrict__ s5,   const float* __restrict__ b5,
    const float* __restrict__ d3a,  const float* __restrict__ s3a,  const float* __restrict__ b3a,
    const float* __restrict__ d3b,  const float* __restrict__ s3b,  const float* __restrict__ b3b,
    const float* __restrict__ d3c,  const float* __restrict__ s3c,  const float* __restrict__ b3c,
    const float* __restrict__ ls,   const float* __restrict__ lb,
    float* __restrict__ effw, float* __restrict__ effb)
{
    int i = blockIdx.x * 256 + threadIdx.x;
    if (i >= CC * K2) return;
    int p = i % K2, c = i / K2;
    int dy = p / 7 - 3, dx = p % 7 - 3;
    float w = lk_s[c] * lk_w[c * 49 + p];
    if (dy >= -2 && dy <= 2 && dx >= -2 && dx <= 2)
        w += s5[c] * dw5[c * 25 + (dy + 2) * 5 + (dx + 2)];
    if (dy >= -1 && dy <= 1 && dx >= -1 && dx <= 1)
        w += s3a[c] * d3a[c * 9 + (dy + 1) * 3 + (dx + 1)];
    if ((dy == -2 || dy == 0 || dy == 2) && (dx == -2 || dx == 0 || dx == 2))
        w += s3b[c] * d3b[c * 9 + (dy / 2 + 1) * 3 + (dx / 2 + 1)];
    if ((dy == -3 || dy == 0 || dy == 3) && (dx == -3 || dx == 0 || dx == 3))
        w += s3c[c] * d3c[c * 9 + (dy / 3 + 1) * 3 + (dx / 3 + 1)];
    effw[i] = ls[c] * w;
    if (p == 0)
        effb[c] = ls[c] * (lk_b[c] + b5[c] + b3a[c] + b3b[c] + b3c[c]) + lb[c];
}

// ---------------------------------------------------------------------------
// 2) Single effective 7x7 depthwise conv -> lepe
// ---------------------------------------------------------------------------
__global__ __launch_bounds__(256) void dw7_lepe_kernel(
    const float* __restrict__ xs, const float* __restrict__ effw,
    const float* __restrict__ effb, float* __restrict__ lepe)
{
    __shared__ float sw[K2];
    int bc = blockIdx.y;            // b*CC + c
    int c  = bc & (CC - 1);
    if (threadIdx.x < K2) sw[threadIdx.x] = effw[c * K2 + threadIdx.x];
    __syncthreads();
    int n = blockIdx.x * 256 + threadIdx.x;
    if (n >= NP) return;
    int h = n / WW, w = n % WW;
    const float* xb = xs + (size_t)bc * NP;
    float acc = effb[c];
#pragma unroll
    for (int dy = -3; dy <= 3; ++dy) {
        int hh = h + dy;
        if (hh < 0 || hh >= HH) continue;
#pragma unroll
        for (int dx = -3; dx <= 3; ++dx) {
            int ww2 = w + dx;
            if (ww2 < 0 || ww2 >= WW) continue;
            acc += sw[(dy + 3) * 7 + (dx + 3)] * xb[hh * WW + ww2];
        }
    }
    lepe[(size_t)bc * NP + n] = acc;
}

// ---------------------------------------------------------------------------
// 3) AdaptiveAvgPool2d(7) on x_up : 8x8 block means -> pooled [B,C,49]
// ---------------------------------------------------------------------------
__global__ void avgpool7_kernel(const float* __restrict__ xup, float* __restrict__ pooled)
{
    int i = blockIdx.x * 256 + threadIdx.x;       // B*C*49
    if (i >= BN * CC * K2) return;
    int t = i % K2, bc = i / K2;
    int ty = t / 7, tx = t % 7;
    const float* p = xup + (size_t)bc * NP + (ty * 8) * WW + tx * 8;
    float s = 0.f;
#pragma unroll
    for (int y = 0; y < 8; ++y)
#pragma unroll
        for (int x = 0; x < 8; ++x) s += p[y * WW + x];
    pooled[i] = s * (1.f / 64.f);
}

// ---------------------------------------------------------------------------
// 4) key = wk @ pooled + bk   (tiny GEMM, plain VALU)
// ---------------------------------------------------------------------------
__global__ void key_proj_kernel(const float* __restrict__ pooled, const float* __restrict__ wk,
                                const float* __restrict__ bk, float* __restrict__ keyb)
{
    int i = blockIdx.x * 256 + threadIdx.x;       // B*C*49
    if (i >= BN * CC * K2) return;
    int t = i % K2; int o = (i / K2) % CC; int b = i / (K2 * CC);
    float s = bk[o];
    const float* pb = pooled + (size_t)b * CC * K2 + t;
    const float* wr = wk + o * CC;
    for (int c = 0; c < CC; ++c) s += wr[c] * pb[c * K2];
    keyb[i] = s;
}

// ---------------------------------------------------------------------------
// 5) 1x1 conv as WMMA f32 GEMM: Y[b,o,n] = sum_c W[o,c] X[b,c,n]  (+epilogue)
//    block = 256 threads = 8 waves; wave w -> M-tile w*16; 4 N-tiles of 16.
//    mode 0: +bias     mode 1: BN + SiLU (gate)     mode 2: +bias + addend
// ---------------------------------------------------------------------------
__global__ __launch_bounds__(256) void conv1x1_wmma_kernel(
    const float* __restrict__ X, const float* __restrict__ W,
    const float* __restrict__ bias, const float* __restrict__ bn_s,
    const float* __restrict__ bn_b, const float* __restrict__ addend,
    float* __restrict__ Y, int mode)
{
    const int b      = blockIdx.y;
    const int n_base = blockIdx.x * 64;           // 49 * 64 == 3136 exactly
    const int wave   = threadIdx.x >> 5;          // 0..7 -> m tile
    const int lane   = threadIdx.x & 31;
    const int m0     = wave * 16;
    const int hi     = lane >> 4;                 // 0|1
    const int lm     = lane & 15;

    const float* Xb = X + (size_t)b * CC * NP;

    v8f acc[4];
#pragma unroll
    for (int nt = 0; nt < 4; ++nt)
#pragma unroll
        for (int r = 0; r < 8; ++r) acc[nt][r] = 0.f;

    for (int k = 0; k < CC; k += 4) {
        // A fragment (16x4 of W): lane (hi,lm) holds W[m0+lm][k+2hi], W[m0+lm][k+2hi+1]
        const float* wp = W + (m0 + lm) * CC + k + 2 * hi;
        v2f a; a.x = wp[0]; a.y = wp[1];
        // B fragments (4x16 of X): rows k+2hi, k+2hi+1 ; cols n_base + nt*16 + lm
        const float* xp0 = Xb + (size_t)(k + 2 * hi) * NP + n_base + lm;
        const float* xp1 = xp0 + NP;
        v2f b0; b0.x = xp0[0];  b0.y = xp1[0];
        v2f b1; b1.x = xp0[16]; b1.y = xp1[16];
        v2f b2; b2.x = xp0[32]; b2.y = xp1[32];
        v2f b3; b3.x = xp0[48]; b3.y = xp1[48];
        acc[0] = __builtin_amdgcn_wmma_f32_16x16x4_f32(false, a, false, b0, (short)0, acc[0], false, false);
        acc[1] = __builtin_amdgcn_wmma_f32_16x16x4_f32(false, a, false, b1, (short)0, acc[1], false, false);
        acc[2] = __builtin_amdgcn_wmma_f32_16x16x4_f32(false, a, false, b2, (short)0, acc[2], false, false);
        acc[3] = __builtin_amdgcn_wmma_f32_16x16x4_f32(false, a, false, b3, (short)0, acc[3], false, false);
    }

#pragma unroll
    for (int nt = 0; nt < 4; ++nt) {
        int n = n_base + nt * 16 + lm;
#pragma unroll
        for (int r = 0; r < 8; ++r) {
            int o = m0 + r + 8 * hi;              // C/D layout: VGPR r -> M = r + 8*hi
            float y = acc[nt][r];
            if (mode == 1) {                      // gate: BN then SiLU
                y = y * bn_s[o] + bn_b[o];
                y = y / (1.f + __expf(-y));
            } else {
                y += bias[o];
                if (mode == 2) y += addend[((size_t)b * CC + o) * NP + n];
            }
            Y[((size_t)b * CC + o) * NP + n] = y;
        }
    }
}

// ---------------------------------------------------------------------------
// 6) Attention: per pixel, 49 logits vs pooled keys, softmax, ww-mix -> dyn_w
// ---------------------------------------------------------------------------
__global__ __launch_bounds__(128) void attn_kernel(
    const float* __restrict__ q, const float* __restrict__ keyb,
    const float* __restrict__ wwg, const float* __restrict__ bwg,
    float* __restrict__ dyn)
{
    __shared__ float skey[CC * K2];
    __shared__ float sww[K2 * K2];
    __shared__ float sbw[K2];
    const int b = blockIdx.y;
    const int tid = threadIdx.x;
    for (int i = tid; i < CC * K2; i += 128) skey[i] = keyb[b * CC * K2 + i];
    for (int i = tid; i < K2 * K2; i += 128) sww[i] = wwg[i];
    if (tid < K2) sbw[tid] = bwg[tid];
    __syncthreads();

    const int n = blockIdx.x * 128 + tid;
    if (n >= NP) return;

    float lg[K2];
#pragma unroll
    for (int t = 0; t < K2; ++t) lg[t] = 0.f;
    const float* qb = q + (size_t)b * CC * NP + n;
    for (int c = 0; c < CC; ++c) {
        float qc = qb[(size_t)c * NP];
#pragma unroll
        for (int t = 0; t < K2; ++t) lg[t] += qc * skey[c * K2 + t];
    }
    const float scale = 0.08838834764831845f;     // 128^-0.5
    float m = -1e30f;
#pragma unroll
    for (int t = 0; t < K2; ++t) { lg[t] *= scale; m = fmaxf(m, lg[t]); }
    float s = 0.f;
#pragma unroll
    for (int t = 0; t < K2; ++t) { lg[t] = __expf(lg[t] - m); s += lg[t]; }
    float inv = 1.f / s;
#pragma unroll
    for (int t = 0; t < K2; ++t) lg[t] *= inv;

    float* dyb = dyn + (size_t)b * K2 * NP + n;
    for (int j = 0; j < K2; ++j) {
        float d = sbw[j];
#pragma unroll
        for (int t = 0; t < K2; ++t) d += sww[j * K2 + t] * lg[t];
        dyb[(size_t)j * NP] = d;
    }
}

// ---------------------------------------------------------------------------
// 7) Dynamic 7x7 mixing of value with per-pixel weights (LDS-tiled, 14x14)
// ---------------------------------------------------------------------------
__global__ __launch_bounds__(196) void mixed_kernel(
    const float* __restrict__ val, const float* __restrict__ dyn,
    float* __restrict__ mixed)
{
    __shared__ float tile[20 * 20];
    const int b = blockIdx.y;
    const int t = blockIdx.x;                     // 16 tiles of 14x14
    const int h0 = (t >> 2) * 14, w0 = (t & 3) * 14;
    const int tid = threadIdx.x;                  // 0..195
    const int ty = tid / 14, tx = tid % 14;
    const int h = h0 + ty, w = w0 + tx;

    float dw[K2];
    const float* dyb = dyn + (size_t)b * K2 * NP + h * WW + w;
#pragma unroll
    for (int j = 0; j < K2; ++j) dw[j] = dyb[(size_t)j * NP];

    for (int c = 0; c < CC; ++c) {
        const float* vb = val + ((size_t)b * CC + c) * NP;
        __syncthreads();
        for (int e = tid; e < 400; e += 196) {
            int lh = e / 20, lw = e % 20;
            int gh = h0 - 3 + lh, gw = w0 - 3 + lw;
            tile[e] = (gh >= 0 && gh < HH && gw >= 0 && gw < WW) ? vb[gh * WW + gw] : 0.f;
        }
        __syncthreads();
        float acc = 0.f;
#pragma unroll
        for (int dy = 0; dy < 7; ++dy)
#pragma unroll
            for (int dx = 0; dx < 7; ++dx)
                acc += dw[dy * 7 + dx] * tile[(ty + dy) * 20 + (tx + dx)];
        mixed[((size_t)b * CC + c) * NP + h * WW + w] = acc;
    }
}

// ---------------------------------------------------------------------------
// 8) Channel-attention reductions: avg & max over HW per (b,c)
// ---------------------------------------------------------------------------
__global__ __launch_bounds__(256) void ca_reduce_kernel(
    const float* __restrict__ xf, float* __restrict__ avg, float* __restrict__ mx)
{
    __shared__ float ssum[256], smax[256];
    const int bc = blockIdx.x;
    const float* p = xf + (size_t)bc * NP;
    float s = 0.f, m = -1e30f;
    for (int i = threadIdx.x; i < NP; i += 256) { float v = p[i]; s += v; m = fmaxf(m, v); }
    ssum[threadIdx.x] = s; smax[threadIdx.x] = m;
    __syncthreads();
    for (int off = 128; off > 0; off >>= 1) {
        if (threadIdx.x < off) {
            ssum[threadIdx.x] += ssum[threadIdx.x + off];
            smax[threadIdx.x] = fmaxf(smax[threadIdx.x], smax[threadIdx.x + off]);
        }
        __syncthreads();
    }
    if (threadIdx.x == 0) { avg[bc] = ssum[0] * (1.f / NP); mx[bc] = smax[0]; }
}

// ---------------------------------------------------------------------------
// 9) Channel-attention MLP: sigmoid(W2 relu(W1 avg) + W2 relu(W1 mx))
// ---------------------------------------------------------------------------
__global__ __launch_bounds__(128) void ca_mlp_kernel(
    const float* __restrict__ avg, const float* __restrict__ mx,
    const float* __restrict__ w1, const float* __restrict__ w2,
    float* __restrict__ ca)
{
    __shared__ float sa[CC], sm[CC], hid[2 * RDIM];
    const int b = blockIdx.x, tid = threadIdx.x;
    sa[tid] = avg[b * CC + tid];
    sm[tid] = mx[b * CC + tid];
    __syncthreads();
    if (tid < 2 * RDIM) {
        int r = tid & (RDIM - 1);
        const float* src = (tid < RDIM) ? sa : sm;
        float acc = 0.f;
        for (int c = 0; c < CC; ++c) acc += w1[r * CC + c] * src[c];
        hid[tid] = fmaxf(acc, 0.f);
    }
    __syncthreads();
    float s = 0.f;
#pragma unroll
    for (int r = 0; r < RDIM; ++r) s += w2[tid * RDIM + r] * (hid[r] + hid[RDIM + r]);
    ca[b * CC + tid] = 1.f / (1.f + __expf(-s));
}

// ---------------------------------------------------------------------------
// 10) out = gate * (ca * x_fused) + x_skip * res_scale
// ---------------------------------------------------------------------------
__global__ __launch_bounds__(256) void final_kernel(
    const float* __restrict__ gate, const float* __restrict__ xf,
    const float* __restrict__ ca, const float* __restrict__ xskip,
    const float* __restrict__ rs, float* __restrict__ out)
{
    size_t i = (size_t)blockIdx.x * 256 + threadIdx.x;
    if (i >= (size_t)BN * CC * NP) return;
    int bc = (int)(i / NP);
    out[i] = gate[i] * (ca[bc] * xf[i]) + xskip[i] * rs[0];
}

// ---------------------------------------------------------------------------
extern "C" void kernel_launch(void* const* d_in, const int* in_sizes, int n_in,
                              void* d_out, int out_size, void* d_ws, size_t ws_size,
                              hipStream_t stream)
{
    (void)in_sizes; (void)n_in; (void)out_size; (void)ws_size;

    const float* x_up   = (const float*)d_in[0];
    const float* x_skip = (const float*)d_in[1];
    const float* wq     = (const float*)d_in[2];
    const float* bq     = (const float*)d_in[3];
    const float* wk     = (const float*)d_in[4];
    const float* bk     = (const float*)d_in[5];
    const float* wv     = (const float*)d_in[6];
    const float* bv     = (const float*)d_in[7];
    const float* ww     = (const float*)d_in[8];
    const float* bw     = (const float*)d_in[9];
    const float* lk_w   = (const float*)d_in[10];
    const float* lk_s   = (const float*)d_in[11];
    const float* lk_b   = (const float*)d_in[12];
    const float* dw5    = (const float*)d_in[13];
    const float* s5     = (const float*)d_in[14];
    const float* b5     = (const float*)d_in[15];
    const float* d3a    = (const float*)d_in[16];
    const float* s3a    = (const float*)d_in[17];
    const float* b3a    = (const float*)d_in[18];
    const float* d3b    = (const float*)d_in[19];
    const float* s3b    = (const float*)d_in[20];
    const float* b3b    = (const float*)d_in[21];
    const float* d3c    = (const float*)d_in[22];
    const float* s3c    = (const float*)d_in[23];
    const float* b3c    = (const float*)d_in[24];
    const float* lps    = (const float*)d_in[25];
    const float* lpb    = (const float*)d_in[26];
    const float* gate_w = (const float*)d_in[27];
    const float* gbn_s  = (const float*)d_in[28];
    const float* gbn_b  = (const float*)d_in[29];
    const float* fus_w  = (const float*)d_in[30];
    const float* fus_b  = (const float*)d_in[31];
    const float* ca_w1  = (const float*)d_in[32];
    const float* ca_w2  = (const float*)d_in[33];
    const float* res_s  = (const float*)d_in[34];

    float* out = (float*)d_out;

    // workspace carve-up
    const size_t BCN = (size_t)BN * CC * NP;      // 1,605,632
    float* ws     = (float*)d_ws;
    float* qbuf   = ws;
    float* vbuf   = qbuf + BCN;
    float* gbuf   = vbuf + BCN;
    float* lepe   = gbuf + BCN;
    float* mixed  = lepe + BCN;
    float* xf     = mixed + BCN;
    float* dyn    = xf + BCN;                     // B*49*NP = 614,656
    float* pooled = dyn + (size_t)BN * K2 * NP;   // 25,088
    float* keyb   = pooled + (size_t)BN * CC * K2;
    float* effw   = keyb + (size_t)BN * CC * K2;  // 6,272
    float* effb   = effw + CC * K2;               // 128
    float* avg    = effb + CC;                    // 512
    float* mxv    = avg + BN * CC;                // 512
    float* ca     = mxv + BN * CC;                // 512

    // 1) fold dilated-reparam branch into one 7x7 kernel
    build_eff_kernel<<<dim3((CC * K2 + 255) / 256), 256, 0, stream>>>(
        lk_w, lk_s, lk_b, dw5, s5, b5, d3a, s3a, b3a, d3b, s3b, b3b,
        d3c, s3c, b3c, lps, lpb, effw, effb);

    // 2) lepe
    dw7_lepe_kernel<<<dim3((NP + 255) / 256, BN * CC), 256, 0, stream>>>(
        x_skip, effw, effb, lepe);

    // 3) pooled keys
    avgpool7_kernel<<<dim3((BN * CC * K2 + 255) / 256), 256, 0, stream>>>(x_up, pooled);
    key_proj_kernel<<<dim3((BN * CC * K2 + 255) / 256), 256, 0, stream>>>(pooled, wk, bk, keyb);

    // 4) WMMA GEMMs: q, v, gate
    conv1x1_wmma_kernel<<<dim3(NP / 64, BN), 256, 0, stream>>>(
        x_skip, wq, bq, nullptr, nullptr, nullptr, qbuf, 0);
    conv1x1_wmma_kernel<<<dim3(NP / 64, BN), 256, 0, stream>>>(
        x_skip, wv, bv, nullptr, nullptr, nullptr, vbuf, 0);
    conv1x1_wmma_kernel<<<dim3(NP / 64, BN), 256, 0, stream>>>(
        x_skip, gate_w, bq /*unused*/, gbn_s, gbn_b, nullptr, gbuf, 1);

    // 5) attention -> dynamic 7x7 weights
    attn_kernel<<<dim3((NP + 127) / 128, BN), 128, 0, stream>>>(qbuf, keyb, ww, bw, dyn);

    // 6) dynamic mixing of value
    mixed_kernel<<<dim3(16, BN), 196, 0, stream>>>(vbuf, dyn, mixed);

    // 7) fusion 1x1 conv + lepe (WMMA, fused epilogue)
    conv1x1_wmma_kernel<<<dim3(NP / 64, BN), 256, 0, stream>>>(
        mixed, fus_w, fus_b, nullptr, nullptr, lepe, xf, 2);

    // 8) channel attention
    ca_reduce_kernel<<<dim3(BN * CC), 256, 0, stream>>>(xf, avg, mxv);
    ca_mlp_kernel<<<dim3(BN), 128, 0, stream>>>(avg, mxv, ca_w1, ca_w2, ca);

    // 9) final gated residual
    final_kernel<<<dim3((unsigned)((BCN + 255) / 256)), 256, 0, stream>>>(
        gbuf, xf, ca, x_skip, res_s, out);
}